// GCN_13786845020199
// MI455X (gfx1250) — compile-verified
//
#include <hip/hip_runtime.h>

typedef float v2f __attribute__((ext_vector_type(2)));
typedef float v8f __attribute__((ext_vector_type(8)));

#define D 64
#define WPAD 66   // LDS row stride (floats) to avoid bank conflicts

// ---- 1. deg[i] = 1.0 (self loop weight) ----
__global__ void gcn_init_deg(float* __restrict__ deg, int n) {
  int i = blockIdx.x * blockDim.x + threadIdx.x;
  if (i < n) deg[i] = 1.0f;
}

// ---- 2. deg[col[e]] += ew[e] ----
__global__ void gcn_deg_accum(float* __restrict__ deg, const int* __restrict__ col,
                              const float* __restrict__ ew, int E) {
  int e = blockIdx.x * blockDim.x + threadIdx.x;
  if (e < E) atomicAdd(&deg[col[e]], ew[e]);
}

// ---- 3. in-place deg -> dis = rsqrt(deg) ----
__global__ void gcn_deg_to_dis(float* __restrict__ deg, int n) {
  int i = blockIdx.x * blockDim.x + threadIdx.x;
  if (i < n) {
    float d = deg[i];
    deg[i] = (d > 0.0f) ? rsqrtf(d) : 0.0f;
  }
}

// ---- 4. h = x @ W^T via V_WMMA_F32_16X16X4_F32; fused: out = h * dis^2 ----
// Block: 128 threads = 4 waves; each wave computes 16 rows x 64 cols.
__global__ void __launch_bounds__(128)
gcn_gemm_wmma(const float* __restrict__ x, const float* __restrict__ W,
              const float* __restrict__ dis, float* __restrict__ h,
              float* __restrict__ out, int n) {
  __shared__ float sW[D * WPAD];
  const int tid = threadIdx.x;
  // stage W (64x64 row-major) into LDS, padded stride
  for (int i = tid; i < D * D; i += 128) {
    sW[(i >> 6) * WPAD + (i & 63)] = W[i];
  }
  __syncthreads();

  const int lane = tid & 31;
  const int wave = tid >> 5;
  const int r0   = blockIdx.x * 64 + wave * 16;  // first row of this wave's tile
  const int m    = lane & 15;
  const int koff = (lane >> 4) << 1;             // 0 (lanes 0-15) or 2 (lanes 16-31)

  int arow = r0 + m;
  if (arow >= n) arow = n - 1;                   // clamp: EXEC must stay all-ones
  const float* __restrict__ xrow = x + (size_t)arow * D;

  v8f acc0 = {}, acc1 = {}, acc2 = {}, acc3 = {};

  for (int kk = 0; kk < D; kk += 4) {
    // A fragment (16x4 fp32): lane<16 holds {K=kk,kk+1}, lane>=16 holds {K=kk+2,kk+3}
    v2f a;
    a.x = xrow[kk + koff];
    a.y = xrow[kk + koff + 1];

    const int c = kk + koff;  // B fragment: Wt[k][col] = W[col][k], col = nt*16 + m
    v2f b0, b1, b2, b3;
    b0.x = sW[(0 * 16 + m) * WPAD + c];  b0.y = sW[(0 * 16 + m) * WPAD + c + 1];
    b1.x = sW[(1 * 16 + m) * WPAD + c];  b1.y = sW[(1 * 16 + m) * WPAD + c + 1];
    b2.x = sW[(2 * 16 + m) * WPAD + c];  b2.y = sW[(2 * 16 + m) * WPAD + c + 1];
    b3.x = sW[(3 * 16 + m) * WPAD + c];  b3.y = sW[(3 * 16 + m) * WPAD + c + 1];

    acc0 = __builtin_amdgcn_wmma_f32_16x16x4_f32(false, a, false, b0, (short)0, acc0, false, false);
    acc1 = __builtin_amdgcn_wmma_f32_16x16x4_f32(false, a, false, b1, (short)0, acc1, false, false);
    acc2 = __builtin_amdgcn_wmma_f32_16x16x4_f32(false, a, false, b2, (short)0, acc2, false, false);
    acc3 = __builtin_amdgcn_wmma_f32_16x16x4_f32(false, a, false, b3, (short)0, acc3, false, false);
  }

  // C/D layout: VGPR v -> M = v (lanes 0-15) or v+8 (lanes 16-31); N = lane&15
  const int nlo = lane & 15;
  const int mhi = (lane >> 4) << 3;  // 0 or 8
  for (int v = 0; v < 8; ++v) {
    int row = r0 + mhi + v;
    if (row < n) {
      float s  = dis[row];
      float ss = s * s;                 // self-loop norm = dis*1*dis
      size_t base = (size_t)row * D;
      h[base +  0 + nlo] = acc0[v];  out[base +  0 + nlo] = acc0[v] * ss;
      h[base + 16 + nlo] = acc1[v];  out[base + 16 + nlo] = acc1[v] * ss;
      h[base + 32 + nlo] = acc2[v];  out[base + 32 + nlo] = acc2[v] * ss;
      h[base + 48 + nlo] = acc3[v];  out[base + 48 + nlo] = acc3[v] * ss;
    }
  }
}

// ---- 5. out[col[e]] += dis[row]*ew*dis[col] * h[row]; 64 threads per edge ----
__global__ void gcn_scatter(const int* __restrict__ ei, const float* __restrict__ ew,
                            const float* __restrict__ dis, const float* __restrict__ h,
                            float* __restrict__ out, int E) {
  unsigned gid = blockIdx.x * blockDim.x + threadIdx.x;
  int e = gid >> 6;
  int f = gid & 63;
  if (e < E) {
    int row = ei[e];
    int col = ei[E + e];
    float nrm = dis[row] * ew[e] * dis[col];
    atomicAdd(&out[(size_t)col * D + f], nrm * h[(size_t)row * D + f]);
  }
}

// ---- 6. out = relu(out + b) ----
__global__ void gcn_finalize(float* __restrict__ out, const float* __restrict__ b, int total) {
  int i = blockIdx.x * blockDim.x + threadIdx.x;
  if (i < total) {
    float v = out[i] + b[i & 63];
    out[i] = v > 0.0f ? v : 0.0f;
  }
}

extern "C" void kernel_launch(void* const* d_in, const int* in_sizes, int n_in,
                              void* d_out, int out_size, void* d_ws, size_t ws_size,
                              hipStream_t stream) {
  const float* x  = (const float*)d_in[0];
  const int*   ei = (const int*)d_in[1];
  const float* ea = (const float*)d_in[2];
  const float* W  = (const float*)d_in[3];
  const float* b  = (const float*)d_in[4];
  float* out = (float*)d_out;

  const int n = in_sizes[0] / D;   // N_NODES
  const int E = in_sizes[2];       // N_EDGES

  // workspace: [deg/dis: n floats][h: n*64 floats]
  float* deg = (float*)d_ws;
  float* h   = deg + ((n + 3) & ~3);

  const int T = 256;
  gcn_init_deg  <<<(n + T - 1) / T, T, 0, stream>>>(deg, n);
  gcn_deg_accum <<<(E + T - 1) / T, T, 0, stream>>>(deg, ei + E, ea, E);
  gcn_deg_to_dis<<<(n + T - 1) / T, T, 0, stream>>>(deg, n);

  gcn_gemm_wmma <<<(n + 63) / 64, 128, 0, stream>>>(x, W, deg, h, out, n);

  size_t sthreads = (size_t)E * 64;
  gcn_scatter   <<<(unsigned)((sthreads + T - 1) / T), T, 0, stream>>>(ei, ea, deg, h, out, E);

  int total = n * D;
  gcn_finalize  <<<(total + T - 1) / T, T, 0, stream>>>(out, b, total);
}